// MultiGaitAdapter_80195629351591
// MI455X (gfx1250) — compile-verified
//
#include <hip/hip_runtime.h>
#include <hip/hip_bf16.h>
#include <math.h>

typedef float v2f __attribute__((ext_vector_type(2)));
typedef float v8f __attribute__((ext_vector_type(8)));

#define LDB ((size_t)4096)

// Workspace layout (float offsets)
constexpr int XT0   = 0;         // 30*25*4096  =  3,072,000
constexpr int H10_0 = 3072000;   // 21*100*4096 =  8,601,600
constexpr int Y0_0  = 11673600;  //  9*100*4096 =  3,686,400
constexpr int H11_0 = 15360000;  //  5*200*4096 =  4,096,000
constexpr int FEAT0 = 19456000;  //    200*4096 =    819,200
constexpr int ZR0   = 20275200;  //        4096 zero row for K-padding

// ---------------------------------------------------------------------------
// Zero the padding row (B-matrix rows for padded K point here)
// ---------------------------------------------------------------------------
__global__ void k_zero(float* __restrict__ ws) {
  ws[ZR0 + blockIdx.x * 256 + threadIdx.x] = 0.f;
}

// ---------------------------------------------------------------------------
// Stage 0: gather x[:, :, 231..255] into XT[c][tw][b]  (b contiguous)
// ---------------------------------------------------------------------------
__global__ void k_gatherx(const float* __restrict__ x, float* __restrict__ ws) {
  __shared__ float tile[64 * 26];
  const int c    = blockIdx.y;
  const int b0   = blockIdx.x * 64;
  const int lane = threadIdx.x & 31;
  const int wv   = threadIdx.x >> 5;
#pragma unroll
  for (int i = 0; i < 8; ++i) {
    const int bsub = wv * 8 + i;
    if (lane < 25)
      tile[bsub * 26 + lane] =
          x[(size_t)(b0 + bsub) * (30 * 256) + (size_t)c * 256 + 231 + lane];
  }
  __syncthreads();
  for (int idx = threadIdx.x; idx < 25 * 64; idx += 256) {
    const int tw = idx >> 6;
    const int bs = idx & 63;
    ws[XT0 + ((size_t)(c * 25 + tw)) * LDB + b0 + bs] = tile[bs * 26 + tw];
  }
}

// ---------------------------------------------------------------------------
// Shared GEMM core: per wave, one 16(M) x 32(N) tile via two WMMA accumulators.
//   lA      : LDS A-tile, layout [r][16] (zero-padded)
//   rowoff  : LDS per-row absolute float offset into ws (padding rows -> ZR0)
// ---------------------------------------------------------------------------
#define GEMM_CORE(KPAD)                                                         \
  v8f acc0 = {0.f, 0.f, 0.f, 0.f, 0.f, 0.f, 0.f, 0.f};                          \
  v8f acc1 = {0.f, 0.f, 0.f, 0.f, 0.f, 0.f, 0.f, 0.f};                          \
  _Pragma("unroll 4")                                                           \
  for (int k = 0; k < (KPAD); k += 4) {                                         \
    const int r0 = k + 2 * half;                                                \
    const int o0 = rowoff[r0];                                                  \
    const int o1 = rowoff[r0 + 1];                                              \
    v2f a, bf0, bf1;                                                            \
    a.x   = lA[r0 * 16 + l15];                                                  \
    a.y   = lA[(r0 + 1) * 16 + l15];                                            \
    bf0.x = ws[(size_t)(o0 + n0)];                                              \
    bf0.y = ws[(size_t)(o1 + n0)];                                              \
    bf1.x = ws[(size_t)(o0 + n1)];                                              \
    bf1.y = ws[(size_t)(o1 + n1)];                                              \
    acc0 = __builtin_amdgcn_wmma_f32_16x16x4_f32(false, a, false, bf0, (short)0,\
                                                 acc0, false, false);           \
    acc1 = __builtin_amdgcn_wmma_f32_16x16x4_f32(false, a, false, bf1, (short)0,\
                                                 acc1, false, false);           \
  }

#define GEMM_PROLOG()                                                           \
  const int lane = threadIdx.x & 31, wv = threadIdx.x >> 5;                     \
  const int half = lane >> 4, l15 = lane & 15;                                  \
  const int n0 = (blockIdx.x * 16 + wv * 2) * 16 + l15;                         \
  const int n1 = n0 + 16;                                                       \
  const int Mb = blockIdx.y * 16;

// ---------------------------------------------------------------------------
// Stage 1: H1_0[tp][m][b] = relu(W1_0(100x150) @ Xcol(150xB) + b1_0), tp=0..20
// ---------------------------------------------------------------------------
__global__ void k_conv1_0(float* __restrict__ ws, const float* __restrict__ w1,
                          const float* __restrict__ b1) {
  __shared__ int   rowoff[152];
  __shared__ float lA[152 * 16];
  const int tp = blockIdx.z;
  GEMM_PROLOG()
  for (int r = threadIdx.x; r < 152; r += 256)
    rowoff[r] = (r < 150) ? XT0 + ((r / 5) * 25 + tp + (r % 5)) * 4096 : ZR0;
  for (int i = threadIdx.x; i < 152 * 16; i += 256) {
    const int r = i >> 4, mm = Mb + (i & 15);
    lA[i] = (mm < 100 && r < 150) ? w1[mm * 150 + r] : 0.f;
  }
  __syncthreads();
  GEMM_CORE(152)
#pragma unroll
  for (int v = 0; v < 8; ++v) {
    const int mm = Mb + v + 8 * half;
    if (mm < 100) {
      const float bv = b1[mm];
      float* orow = ws + H10_0 + ((size_t)tp * 100 + mm) * LDB;
      orow[n0] = fmaxf(acc0[v] + bv, 0.f);
      orow[n1] = fmaxf(acc1[v] + bv, 0.f);
    }
  }
}

// ---------------------------------------------------------------------------
// Stage 2: Y0[tq][m][b] = relu(W2_0@H1col + WD0@Xpt + b2 + bd), K=500+30, tq=0..8
// ---------------------------------------------------------------------------
__global__ void k_conv2_0(float* __restrict__ ws, const float* __restrict__ w2,
                          const float* __restrict__ b2, const float* __restrict__ wd,
                          const float* __restrict__ bd) {
  __shared__ int   rowoff[532];
  __shared__ float lA[532 * 16];
  const int tq = blockIdx.z;
  GEMM_PROLOG()
  for (int r = threadIdx.x; r < 532; r += 256)
    rowoff[r] = (r < 500) ? H10_0 + ((2 * tq + r % 5) * 100 + r / 5) * 4096
              : (r < 530) ? XT0 + ((r - 500) * 25 + 8 + 2 * tq) * 4096
                          : ZR0;
  for (int i = threadIdx.x; i < 532 * 16; i += 256) {
    const int r = i >> 4, mm = Mb + (i & 15);
    lA[i] = (mm >= 100) ? 0.f
          : (r < 500)   ? w2[mm * 500 + r]
          : (r < 530)   ? wd[mm * 30 + (r - 500)]
                        : 0.f;
  }
  __syncthreads();
  GEMM_CORE(532)
#pragma unroll
  for (int v = 0; v < 8; ++v) {
    const int mm = Mb + v + 8 * half;
    if (mm < 100) {
      const float bv = b2[mm] + bd[mm];
      float* orow = ws + Y0_0 + ((size_t)tq * 100 + mm) * LDB;
      orow[n0] = fmaxf(acc0[v] + bv, 0.f);
      orow[n1] = fmaxf(acc1[v] + bv, 0.f);
    }
  }
}

// ---------------------------------------------------------------------------
// Stage 3: H1_1[tu][m][b] = relu(W1_1(200x500) @ Y0col + b1_1), tu=0..4
// ---------------------------------------------------------------------------
__global__ void k_conv1_1(float* __restrict__ ws, const float* __restrict__ w1,
                          const float* __restrict__ b1) {
  __shared__ int   rowoff[500];
  __shared__ float lA[500 * 16];
  const int tu = blockIdx.z;
  GEMM_PROLOG()
  for (int r = threadIdx.x; r < 500; r += 256)
    rowoff[r] = Y0_0 + ((tu + r % 5) * 100 + r / 5) * 4096;
  for (int i = threadIdx.x; i < 500 * 16; i += 256) {
    const int r = i >> 4, mm = Mb + (i & 15);
    lA[i] = (mm < 200) ? w1[mm * 500 + r] : 0.f;
  }
  __syncthreads();
  GEMM_CORE(500)
#pragma unroll
  for (int v = 0; v < 8; ++v) {
    const int mm = Mb + v + 8 * half;
    if (mm < 200) {
      const float bv = b1[mm];
      float* orow = ws + H11_0 + ((size_t)tu * 200 + mm) * LDB;
      orow[n0] = fmaxf(acc0[v] + bv, 0.f);
      orow[n1] = fmaxf(acc1[v] + bv, 0.f);
    }
  }
}

// ---------------------------------------------------------------------------
// Stage 4: FEAT[m][b] = relu(W2_1@H11col + WD1@Y0[8] + b2 + bd), K=1000+100
// ---------------------------------------------------------------------------
__global__ void k_conv2_1(float* __restrict__ ws, const float* __restrict__ w2,
                          const float* __restrict__ b2, const float* __restrict__ wd,
                          const float* __restrict__ bd) {
  __shared__ int   rowoff[1100];
  __shared__ float lA[1100 * 16];
  GEMM_PROLOG()
  for (int r = threadIdx.x; r < 1100; r += 256)
    rowoff[r] = (r < 1000) ? H11_0 + ((r % 5) * 200 + r / 5) * 4096
                           : Y0_0 + (800 + (r - 1000)) * 4096;
  for (int i = threadIdx.x; i < 1100 * 16; i += 256) {
    const int r = i >> 4, mm = Mb + (i & 15);
    lA[i] = (mm >= 200) ? 0.f
          : (r < 1000)  ? w2[mm * 1000 + r]
                        : wd[mm * 100 + (r - 1000)];
  }
  __syncthreads();
  GEMM_CORE(1100)
#pragma unroll
  for (int v = 0; v < 8; ++v) {
    const int mm = Mb + v + 8 * half;
    if (mm < 200) {
      const float bv = b2[mm] + bd[mm];
      float* orow = ws + FEAT0 + (size_t)mm * LDB;
      orow[n0] = fmaxf(acc0[v] + bv, 0.f);
      orow[n1] = fmaxf(acc1[v] + bv, 0.f);
    }
  }
}

// ---------------------------------------------------------------------------
// Stage 5: per-sample task head; one wave32 per batch element
// ---------------------------------------------------------------------------
__global__ void k_head(const float* __restrict__ ws, const int* __restrict__ task,
                       const float* __restrict__ hw1, const float* __restrict__ hb1,
                       const float* __restrict__ hw2, const float* __restrict__ hb2,
                       float* __restrict__ out) {
  const int lane = threadIdx.x & 31;
  const int wv   = threadIdx.x >> 5;
  const int b    = blockIdx.x * 8 + wv;
  const int h    = task[b];
  const float* feat = ws + FEAT0;
  const float* W1   = hw1 + (size_t)h * 100 * 200;
  const int o0 = lane, o1 = lane + 32, o2 = lane + 64, o3 = lane + 96;
  float s0 = 0.f, s1 = 0.f, s2 = 0.f, s3 = 0.f;
  for (int d = 0; d < 200; ++d) {
    const float f = feat[(size_t)d * LDB + b];
    s0 += W1[o0 * 200 + d] * f;
    s1 += W1[o1 * 200 + d] * f;
    s2 += W1[o2 * 200 + d] * f;
    if (lane < 4) s3 += W1[o3 * 200 + d] * f;
  }
  const float* B1 = hb1 + h * 100;
  const float* W2 = hw2 + h * 100;  // hw2: (H, 1, 100)
  float p = tanhf(s0 + B1[o0]) * W2[o0] + tanhf(s1 + B1[o1]) * W2[o1] +
            tanhf(s2 + B1[o2]) * W2[o2];
  if (lane < 4) p += tanhf(s3 + B1[o3]) * W2[o3];
#pragma unroll
  for (int off = 16; off > 0; off >>= 1) p += __shfl_xor(p, off, 32);
  if (lane == 0) out[b] = p + hb2[h];
}

// ---------------------------------------------------------------------------
extern "C" void kernel_launch(void* const* d_in, const int* in_sizes, int n_in,
                              void* d_out, int out_size, void* d_ws, size_t ws_size,
                              hipStream_t stream) {
  const float* x    = (const float*)d_in[0];
  const int*   task = (const int*)d_in[1];
  const float* w1_0 = (const float*)d_in[2];
  const float* b1_0 = (const float*)d_in[3];
  const float* w2_0 = (const float*)d_in[4];
  const float* b2_0 = (const float*)d_in[5];
  const float* wd_0 = (const float*)d_in[6];
  const float* bd_0 = (const float*)d_in[7];
  const float* w1_1 = (const float*)d_in[8];
  const float* b1_1 = (const float*)d_in[9];
  const float* w2_1 = (const float*)d_in[10];
  const float* b2_1 = (const float*)d_in[11];
  const float* wd_1 = (const float*)d_in[12];
  const float* bd_1 = (const float*)d_in[13];
  const float* hw1  = (const float*)d_in[14];
  const float* hb1  = (const float*)d_in[15];
  const float* hw2  = (const float*)d_in[16];
  const float* hb2  = (const float*)d_in[17];

  float* ws = (float*)d_ws;

  k_zero<<<16, 256, 0, stream>>>(ws);
  k_gatherx<<<dim3(64, 30), 256, 0, stream>>>(x, ws);
  k_conv1_0<<<dim3(16, 7, 21), 256, 0, stream>>>(ws, w1_0, b1_0);
  k_conv2_0<<<dim3(16, 7, 9), 256, 0, stream>>>(ws, w2_0, b2_0, wd_0, bd_0);
  k_conv1_1<<<dim3(16, 13, 5), 256, 0, stream>>>(ws, w1_1, b1_1);
  k_conv2_1<<<dim3(16, 13, 1), 256, 0, stream>>>(ws, w2_1, b2_1, wd_1, bd_1);
  k_head<<<512, 256, 0, stream>>>(ws, task, hw1, hb1, hw2, hb2, (float*)d_out);
}